// MaXDeepLabLoss_54752243089788
// MI455X (gfx1250) — compile-verified
//
#include <hip/hip_runtime.h>
#include <math.h>

typedef __attribute__((ext_vector_type(2))) float v2f;
typedef __attribute__((ext_vector_type(8))) float v8f;

#define HW_PIX 25921
#define NB 8
#define NM 128
#define NG 64
#define NC 133      // num classes (pos logits 0..132, neg index 133)
#define NCP1 134    // class-logit width
#define NEGV -999.0f
#define EPSV 1e-5f
#define ALPHAV 0.75f

// ---------------- kernel 0: zero the inter accumulator ----------------
__global__ void k_zero(float* __restrict__ p, int n) {
  int i = blockIdx.x * blockDim.x + threadIdx.x;
  if (i < n) p[i] = 0.f;
}

// ---- kernel 1: fused per-pixel softmax(M=128) + segmented accumulation ----
// inter[b,g,m] += pred[b,p,m] for g = gt[b,p].  LDS 64x128 accumulator per
// block, ds_add_f32 within block, global atomic f32 flush.  Reads the 106MB
// logits tensor exactly once (memory-bound roofline ~4.6us @ 23.3 TB/s).
__global__ __launch_bounds__(256) void k_softmax_scatter(
    const float* __restrict__ logits, const int* __restrict__ gta,
    float* __restrict__ inter) {
  __shared__ float acc[NG * NM];  // 32 KB
  const int b = blockIdx.y;
  const int tid = threadIdx.x, lane = tid & 31, wave = tid >> 5;
  for (int i = tid; i < NG * NM; i += 256) acc[i] = 0.f;
  __syncthreads();

  const int nb = gridDim.x;
  const int per = (HW_PIX + nb - 1) / nb;
  const int p0 = blockIdx.x * per;
  const int p1 = (p0 + per < HW_PIX) ? (p0 + per) : HW_PIX;
  const float* base = logits + (size_t)b * HW_PIX * NM;
  const int* gb = gta + (size_t)b * HW_PIX;

  for (int p = p0 + wave; p < p1; p += 8) {
    const float* row = base + (size_t)p * NM;
    __builtin_prefetch(row + 8 * NM, 0, 1);  // global_prefetch_b8 for next pixel
    // lane owns m = lane, lane+32, lane+64, lane+96 (bank-conflict-free LDS)
    float v0 = row[lane], v1 = row[lane + 32], v2 = row[lane + 64], v3 = row[lane + 96];
    float mx = fmaxf(fmaxf(v0, v1), fmaxf(v2, v3));
    #pragma unroll
    for (int off = 16; off; off >>= 1) mx = fmaxf(mx, __shfl_xor(mx, off, 32));
    float e0 = expf(v0 - mx), e1 = expf(v1 - mx), e2 = expf(v2 - mx), e3 = expf(v3 - mx);
    float s = (e0 + e1) + (e2 + e3);
    #pragma unroll
    for (int off = 16; off; off >>= 1) s += __shfl_xor(s, off, 32);
    float inv = 1.f / s;
    int g = gb[p];
    float* a = acc + g * NM + lane;
    atomicAdd(a,       e0 * inv);
    atomicAdd(a + 32,  e1 * inv);
    atomicAdd(a + 64,  e2 * inv);
    atomicAdd(a + 96,  e3 * inv);
  }
  __syncthreads();
  float* out = inter + (size_t)b * NG * NM;
  for (int i = tid; i < NG * NM; i += 256) atomicAdd(&out[i], acc[i]);
}

// ---- kernel 2: class softmax + log-softmax over 134, one wave per row ----
__global__ __launch_bounds__(256) void k_class_softmax(
    const float* __restrict__ x, float* __restrict__ prob, float* __restrict__ logp) {
  int row = blockIdx.x * 8 + (threadIdx.x >> 5);  // [0, 1024)
  int lane = threadIdx.x & 31;
  const float* xr = x + (size_t)row * NCP1;
  float v[5];
  float mx = -1e30f;
  #pragma unroll
  for (int t = 0; t < 5; ++t) {
    int c = lane + 32 * t;
    v[t] = (c < NCP1) ? xr[c] : -1e30f;
    mx = fmaxf(mx, v[t]);
  }
  #pragma unroll
  for (int off = 16; off; off >>= 1) mx = fmaxf(mx, __shfl_xor(mx, off, 32));
  float e[5];
  float s = 0.f;
  #pragma unroll
  for (int t = 0; t < 5; ++t) {
    e[t] = expf(v[t] - mx);
    if (lane + 32 * t < NCP1) s += e[t];
  }
  #pragma unroll
  for (int off = 16; off; off >>= 1) s += __shfl_xor(s, off, 32);
  float inv = 1.f / s, ls = logf(s);
  #pragma unroll
  for (int t = 0; t < 5; ++t) {
    int c = lane + 32 * t;
    if (c < NCP1) {
      prob[(size_t)row * NCP1 + c] = e[t] * inv;
      logp[(size_t)row * NCP1 + c] = (v[t] - mx) - ls;
    }
  }
}

// ---- kernel 3: cls_sim = sem_onehot(64x134) x class_prob^T(134x128) via WMMA ----
// Exact f32 GEMM with V_WMMA_F32_16X16X4_F32. One wave per 16x16 output tile.
// Operand layouts per ISA 7.12.2: A/B component j <-> K = k0 + j + 2*(lane>=16);
// A row = lane&15 (both halves); B col = lane&15; D vgpr j -> row j + 8*(lane>=16).
// Main loop k0=0..128 needs no bounds checks (k <= 131 < 134); a single guarded
// tail step covers k=132..135 so the hot loop has unpredicated b64 loads.
__global__ __launch_bounds__(32) void k_cls_sim_wmma(
    const float* __restrict__ class_prob, const int* __restrict__ sem,
    float* __restrict__ cls_sim) {
  const int mt = blockIdx.x;   // 0..7  (M tile)
  const int gt = blockIdx.y;   // 0..3  (G tile)
  const int b  = blockIdx.z;   // 0..7
  const int lane = threadIdx.x;
  const int lo = lane & 15, hi = lane >> 4;

  const int g_a = gt * 16 + lo;
  int sg = sem[b * NG + g_a];
  const int sem_safe = (sg < 0) ? 0 : sg;         // reference uses max(sem,0)
  const int mcol = mt * 16 + lo;
  const float* cp = class_prob + ((size_t)b * NM + mcol) * NCP1;

  v8f acc = {};
  // 33 full K-steps, unconditional loads.  cp[ka] is 8B aligned (ka and the
  // row offset (b*128+m)*134 are both even), so one b64 load per operand.
  for (int k0 = 0; k0 <= 128; k0 += 4) {
    const int ka = k0 + 2 * hi;                   // component .x ; .y = ka+1
    v2f a;
    a.x = (sem_safe == ka)     ? 1.0f : 0.0f;
    a.y = (sem_safe == ka + 1) ? 1.0f : 0.0f;
    v2f bb = *(const v2f*)(cp + ka);
    acc = __builtin_amdgcn_wmma_f32_16x16x4_f32(
        false, a, false, bb, (short)0, acc, false, false);
  }
  {  // tail: k0 = 132, k in [132, 135], guard the two loads
    const int ka = 132 + 2 * hi;
    v2f a, bb;
    a.x = (sem_safe == ka)     ? 1.0f : 0.0f;
    a.y = (sem_safe == ka + 1) ? 1.0f : 0.0f;
    bb.x = (ka     < NCP1) ? cp[ka]     : 0.0f;
    bb.y = (ka + 1 < NCP1) ? cp[ka + 1] : 0.0f;
    acc = __builtin_amdgcn_wmma_f32_16x16x4_f32(
        false, a, false, bb, (short)0, acc, false, false);
  }
  float* out = cls_sim + (size_t)b * NG * NM + (size_t)(gt * 16 + 8 * hi) * NM + mt * 16 + lo;
  #pragma unroll
  for (int j = 0; j < 8; ++j) out[(size_t)j * NM] = acc[j];
}

// ---- kernel 4: dice + masked similarity.  gt_count/colsum derived from inter ----
__global__ __launch_bounds__(256) void k_dice_sim(
    const float* __restrict__ inter, const float* __restrict__ csim,
    const int* __restrict__ sem, float* __restrict__ dice, float* __restrict__ sim) {
  __shared__ float sh[NG * NM];
  __shared__ float gc[NG];
  __shared__ float cs[NM];
  const int b = blockIdx.x, tid = threadIdx.x;
  const float* in = inter + (size_t)b * NG * NM;
  for (int i = tid; i < NG * NM; i += 256) sh[i] = in[i];
  __syncthreads();
  if (tid < NG) {                       // gt pixel count = row sum (softmax rows sum to 1)
    float s = 0.f;
    for (int m = 0; m < NM; ++m) s += sh[tid * NM + m];
    gc[tid] = s;
  } else if (tid < NG + NM) {           // pred column sum = col sum
    int m = tid - NG;
    float s = 0.f;
    for (int g = 0; g < NG; ++g) s += sh[g * NM + m];
    cs[m] = s;
  }
  __syncthreads();
  for (int i = tid; i < NG * NM; i += 256) {
    int g = i >> 7, m = i & 127;
    float d = sh[i] / (0.5f * (gc[g] + cs[m]) + EPSV);
    dice[(size_t)b * NG * NM + i] = d;
    int sg = sem[b * NG + g];
    sim[(size_t)b * NG * NM + i] = (sg >= 0) ? csim[(size_t)b * NG * NM + i] * d : NEGV;
  }
}

// ---- kernel 5: Jonker-Volgenant Hungarian (min-cost), one wave per batch ----
// n = 128 padded (rows >= 64 cost 999 = -NEG).  f64 to mirror the numpy
// reference; first-index tie-break in argmin like np.argmin.
// Lane owns columns j = lane+1+32t (t=0..3): v/minv/used in registers,
// u/p/way in LDS.
#define HN 128
__global__ __launch_bounds__(32) void k_hungarian(
    const float* __restrict__ sim, int* __restrict__ matched) {
  const int b = blockIdx.x;
  const int lane = threadIdx.x;
  __shared__ double u_lds[HN + 1];
  __shared__ int p_lds[HN + 1];
  __shared__ int way_lds[HN + 1];
  const float* S = sim + (size_t)b * NG * NM;
  const double INFD = 1e18;

  double v_loc[4];
  #pragma unroll
  for (int t = 0; t < 4; ++t) v_loc[t] = 0.0;
  for (int j = lane; j <= HN; j += 32) { u_lds[j] = 0.0; p_lds[j] = 0; way_lds[j] = 0; }
  __syncthreads();

  for (int i = 1; i <= HN; ++i) {
    if (lane == 0) p_lds[0] = i;
    unsigned usedmask = 0;
    bool used0 = false;
    double minv_l[4];
    #pragma unroll
    for (int t = 0; t < 4; ++t) minv_l[t] = INFD;
    int j0 = 0;
    __syncthreads();

    while (true) {
      // mark used[j0]
      if (j0 == 0) used0 = true;
      else if (((j0 - 1) & 31) == lane) usedmask |= 1u << ((j0 - 1) >> 5);
      const int i0 = p_lds[j0];
      const double u_i0 = u_lds[i0];
      const int r = i0 - 1;
      // relax free columns owned by this lane
      #pragma unroll
      for (int t = 0; t < 4; ++t) {
        if (!(usedmask & (1u << t))) {
          const int j = lane + 1 + 32 * t;
          double c = (r < NG) ? -(double)S[r * NM + (j - 1)] : 999.0;
          double cur = c - u_i0 - v_loc[t];
          if (cur < minv_l[t]) { minv_l[t] = cur; way_lds[j] = j0; }
        }
      }
      // argmin over free columns, first-index tie-break
      double bv = INFD;
      int bj = HN + 1;
      #pragma unroll
      for (int t = 0; t < 4; ++t) {
        if (!(usedmask & (1u << t))) {
          const int j = lane + 1 + 32 * t;
          if (minv_l[t] < bv) { bv = minv_l[t]; bj = j; }
        }
      }
      #pragma unroll
      for (int off = 16; off; off >>= 1) {
        double ov = __shfl_xor(bv, off, 32);
        int    oj = __shfl_xor(bj, off, 32);
        if (ov < bv || (ov == bv && oj < bj)) { bv = ov; bj = oj; }
      }
      const double delta = bv;
      const int j1 = bj;
      // dual updates
      if (used0 && lane == 0) u_lds[p_lds[0]] += delta;   // u[i] += delta
      #pragma unroll
      for (int t = 0; t < 4; ++t) {
        const int j = lane + 1 + 32 * t;
        if (usedmask & (1u << t)) {
          u_lds[p_lds[j]] += delta;   // distinct rows per used column -> race-free
          v_loc[t] -= delta;
        } else {
          minv_l[t] -= delta;
        }
      }
      j0 = j1;
      __syncthreads();
      if (p_lds[j0] == 0) break;
    }
    // augment along the alternating path
    if (lane == 0) {
      int jj = j0;
      while (jj != 0) {
        int jn = way_lds[jj];
        p_lds[jj] = p_lds[jn];
        jj = jn;
      }
    }
    __syncthreads();
  }
  // col_for_row: p[j] = row assigned to column j
  for (int j = lane + 1; j <= HN; j += 32) {
    int row = p_lds[j];
    matched[b * HN + (row - 1)] = j - 1;
  }
}

// ---- kernel 6: final scalar loss.  Matched slots form a permutation, so the
// reference's scatter-adds reduce to an inverse-map gather. ----
__global__ __launch_bounds__(256) void k_finalize(
    const float* __restrict__ dice, const float* __restrict__ csim,
    const float* __restrict__ logp, const int* __restrict__ sem,
    const int* __restrict__ matched, float* __restrict__ out) {
  __shared__ int gos[NB * NM];
  __shared__ float r0[256], r1[256], r2[256];
  const int tid = threadIdx.x;
  for (int i = tid; i < NB * NM; i += 256) gos[i] = -1;
  __syncthreads();
  for (int i = tid; i < NB * NG; i += 256) {
    int b = i >> 6, g = i & 63;
    gos[b * NM + matched[b * NM + g]] = g;
  }
  __syncthreads();
  float ca = 0.f, da = 0.f, nv = 0.f;
  // class loss over (b, m)
  for (int i = tid; i < NB * NM; i += 256) {
    int b = i >> 7, m = i & 127;
    const float* lp = logp + (size_t)i * NCP1;
    int g = gos[i];
    int sg = (g >= 0) ? sem[b * NG + g] : -1;
    if (sg >= 0) {
      float dps = dice[((size_t)b * NG + g) * NM + m];  // matched_dice scattered to slot m
      ca += ALPHAV * dps * (-lp[sg]);                   // alpha * dice_per_slot * pos_ce
    } else {
      ca += (1.0f - ALPHAV) * (-lp[NC]);                // (1-alpha) * neg_ce
    }
  }
  // dice loss over valid (b, g)
  for (int i = tid; i < NB * NG; i += 256) {
    int sg = sem[i];
    if (sg >= 0) {
      int b = i >> 6, g = i & 63;
      int s = matched[b * NM + g];
      float md = dice[((size_t)b * NG + g) * NM + s];
      float mc = csim[((size_t)b * NG + g) * NM + s];
      da += -mc * logf(md + EPSV);
      nv += 1.f;
    }
  }
  r0[tid] = ca; r1[tid] = da; r2[tid] = nv;
  __syncthreads();
  for (int s = 128; s; s >>= 1) {
    if (tid < s) { r0[tid] += r0[tid + s]; r1[tid] += r1[tid + s]; r2[tid] += r2[tid + s]; }
    __syncthreads();
  }
  if (tid == 0) out[0] = r0[0] / (float)(NB * NM) + r1[0] / (r2[0] + 1.f);
}

extern "C" void kernel_launch(void* const* d_in, const int* in_sizes, int n_in,
                              void* d_out, int out_size, void* d_ws, size_t ws_size,
                              hipStream_t stream) {
  (void)in_sizes; (void)n_in; (void)out_size; (void)ws_size;
  const float* logits = (const float*)d_in[0];   // (8, 25921, 128) f32
  const float* clslog = (const float*)d_in[1];   // (8, 128, 134)   f32
  const int*   gta    = (const int*)d_in[2];     // (8, 25921)      i32
  const int*   sem    = (const int*)d_in[3];     // (8, 64)         i32

  float* ws    = (float*)d_ws;
  float* inter = ws;                                   // 8*64*128
  float* prob  = inter + (size_t)NB * NG * NM;         // 8*128*134
  float* logp  = prob  + (size_t)NB * NM * NCP1;       // 8*128*134
  float* csim  = logp  + (size_t)NB * NM * NCP1;       // 8*64*128
  float* dice  = csim  + (size_t)NB * NG * NM;         // 8*64*128
  float* sim   = dice  + (size_t)NB * NG * NM;         // 8*64*128
  int*   match = (int*)(sim + (size_t)NB * NG * NM);   // 8*128  (~2.2 MB total)

  const int n_inter = NB * NG * NM;
  k_zero<<<dim3((n_inter + 255) / 256), 256, 0, stream>>>(inter, n_inter);
  k_softmax_scatter<<<dim3(64, NB), 256, 0, stream>>>(logits, gta, inter);
  k_class_softmax<<<dim3(NB * NM / 8), 256, 0, stream>>>(clslog, prob, logp);
  k_cls_sim_wmma<<<dim3(NM / 16, NG / 16, NB), 32, 0, stream>>>(prob, sem, csim);
  k_dice_sim<<<dim3(NB), 256, 0, stream>>>(inter, csim, sem, dice, sim);
  k_hungarian<<<dim3(NB), 32, 0, stream>>>(sim, match);
  k_finalize<<<dim3(1), 256, 0, stream>>>(dice, csim, logp, sem, match, (float*)d_out);
}